// MemoryEnhancedGeoSegformer_5549097747010
// MI455X (gfx1250) — compile-verified
//
#include <hip/hip_runtime.h>
#include <hip/hip_bf16.h>

// ---------------------------------------------------------------------------
// MemoryEnhancedGeoSegformer for MI455X (gfx1250), compile-only optimized.
//
// Key algebraic simplification: attention has kv_len==1, so
// softmax(scores, axis=-1) == 1.0 exactly -> Q/wq/bq/wk/bk are dead code and
// the fused output is  out[b,d,h,w] = img[b,d,h,w] + V[b,d],
// with V = loc @ wv + bv. The job is then:
//   * tiny M=16 GEMMs  -> V_WMMA_F32_16X16X4_F32 (full fp32 precision)
//   * one 604 MB streaming broadcast-add -> pure HBM-bandwidth kernel (~26us)
// ---------------------------------------------------------------------------

typedef __attribute__((ext_vector_type(2))) float v2f;
typedef __attribute__((ext_vector_type(8))) float v8f;

// ---------------------------------------------------------------------------
// Kernel 1: multi-scale random Fourier features.  r[b, 0:512] =
//   [cos(y0) sin(y0) cos(y1) sin(y1) cos(y2) sin(y2)], y_i = gps @ om_i^T + b_i
// grid = B(16), block = 512.
// ---------------------------------------------------------------------------
__global__ void rff_kernel(const float* __restrict__ gps,
                           const float* __restrict__ om0, const float* __restrict__ b0,
                           const float* __restrict__ om1, const float* __restrict__ b1,
                           const float* __restrict__ om2, const float* __restrict__ b2,
                           float* __restrict__ r) {
    const int b = blockIdx.x;
    const int j = threadIdx.x;             // 0..511
    const float gx = gps[2 * b + 0];
    const float gy = gps[2 * b + 1];

    const float* om;  const float* bb;  int f;  bool is_sin;
    if (j < 172) {                          // scale 0: 86 cos + 86 sin
        om = om0; bb = b0;
        is_sin = (j >= 86); f = is_sin ? (j - 86) : j;
    } else if (j < 342) {                   // scale 1: 85 cos + 85 sin
        om = om1; bb = b1;
        int t = j - 172; is_sin = (t >= 85); f = is_sin ? (t - 85) : t;
    } else {                                // scale 2: 85 cos + 85 sin
        om = om2; bb = b2;
        int t = j - 342; is_sin = (t >= 85); f = is_sin ? (t - 85) : t;
    }
    const float y = gx * om[2 * f + 0] + gy * om[2 * f + 1] + bb[f];
    r[b * 512 + j] = is_sin ? sinf(y) : cosf(y);
}

// ---------------------------------------------------------------------------
// Kernel 2: M=16 GEMM with WMMA f32 16x16x4.  C[16,N] = act(A[16,K] @ B[K,N] + bias)
// One wave (32 threads) per 16-wide tile of N; K stepped by 4.
// Fragment layouts per CDNA5 ISA 7.12.2:
//   A 16x4 : lanes 0-15 -> M=lane, {K=k0,k0+1}; lanes 16-31 -> M=lane-16, {K=k0+2,k0+3}
//   B 4x16 : lanes 0-15 -> N=lane, rows {k0,k0+1}; lanes 16-31 -> rows {k0+2,k0+3}
//   C/D    : VGPR i, lanes 0-15 -> M=i, N=lane; lanes 16-31 -> M=i+8, N=lane-16
// ---------------------------------------------------------------------------
__global__ void gemm16_wmma(const float* __restrict__ A,
                            const float* __restrict__ Bw,
                            const float* __restrict__ bias,
                            float* __restrict__ C,
                            int K, int N, int relu_flag) {
    const int lane = threadIdx.x;          // 0..31, one full wave (EXEC all 1s)
    const int half = lane >> 4;            // 0 or 1
    const int m    = lane & 15;
    const int ncol = (blockIdx.x << 4) + (lane & 15);

    v8f c = {};
    for (int k0 = 0; k0 < K; k0 += 4) {
        const int ka = k0 + (half << 1);
        v2f a, b;
        a.x = A[m * K + ka];
        a.y = A[m * K + ka + 1];
        b.x = Bw[(size_t)ka * N + ncol];
        b.y = Bw[(size_t)(ka + 1) * N + ncol];
        c = __builtin_amdgcn_wmma_f32_16x16x4_f32(
                /*neg_a=*/false, a, /*neg_b=*/false, b,
                /*c_mod=*/(short)0, c, /*reuse_a=*/false, /*reuse_b=*/false);
    }

    const float bv = bias[ncol];
#pragma unroll
    for (int i = 0; i < 8; ++i) {
        const int row = i + (half << 3);
        float v = c[i] + bv;
        if (relu_flag) v = fmaxf(v, 0.0f);
        C[row * N + ncol] = v;
    }
}

// ---------------------------------------------------------------------------
// Kernel 3: LayerNorm + memory-bank gather-mean + 0.5 blend.
//   locf[b,d] = 0.5*( LN(loc[b,:])[d]*g[d]+beta[d] + mean_k membank[idx[b],k,d] )
// grid = 16, block = 512 (one row per block).
// ---------------------------------------------------------------------------
__global__ void ln_blend(const float* __restrict__ loc,
                         const float* __restrict__ ln_g,
                         const float* __restrict__ ln_b,
                         const float* __restrict__ membank,
                         const int*   __restrict__ mem_idx,
                         float* __restrict__ locf) {
    __shared__ float red[512];
    const int b = blockIdx.x;
    const int d = threadIdx.x;
    const float x = loc[b * 512 + d];

    // mean
    red[d] = x;
    __syncthreads();
    for (int s = 256; s > 0; s >>= 1) {
        if (d < s) red[d] += red[d + s];
        __syncthreads();
    }
    const float mu = red[0] * (1.0f / 512.0f);
    __syncthreads();

    // variance (biased, matches jnp.var)
    const float dv = x - mu;
    red[d] = dv * dv;
    __syncthreads();
    for (int s = 256; s > 0; s >>= 1) {
        if (d < s) red[d] += red[d + s];
        __syncthreads();
    }
    const float var = red[0] * (1.0f / 512.0f);

    // memory bank gather + mean over K=20 slots
    const size_t base = (size_t)mem_idx[b] * 20 * 512 + d;
    float mem = 0.0f;
#pragma unroll
    for (int k = 0; k < 20; ++k) mem += membank[base + (size_t)k * 512];
    mem *= (1.0f / 20.0f);

    const float ln = dv * rsqrtf(var + 1e-5f) * ln_g[d] + ln_b[d];
    locf[b * 512 + d] = 0.5f * (ln + mem);
}

// ---------------------------------------------------------------------------
// Kernel 4: the bandwidth-bound part.  out[b,d,:,:] = img[b,d,:,:] + V[b,d]
// One block per (b,d) row (9216 contiguous floats = 2304 float4).
// grid = 16*512 = 8192, block = 256.
// ---------------------------------------------------------------------------
__global__ void fuse_add(const float* __restrict__ img,
                         const float* __restrict__ V,
                         float* __restrict__ out) {
    const size_t row = blockIdx.x;               // b*512 + d
    const float vv = V[row];
    const float4* __restrict__ in4  = reinterpret_cast<const float4*>(img + row * 9216);
    float4* __restrict__       out4 = reinterpret_cast<float4*>(out + row * 9216);
#pragma unroll
    for (int it = 0; it < 9; ++it) {             // 9 * 256 = 2304 float4
        const int i = it * 256 + threadIdx.x;
        // stream-prefetch the next chunk (emits global_prefetch_b8 on gfx1250)
        if (it < 8) __builtin_prefetch(&in4[i + 256], 0, 0);
        float4 t = in4[i];
        t.x += vv; t.y += vv; t.z += vv; t.w += vv;
        out4[i] = t;
    }
}

// ---------------------------------------------------------------------------
// Host-side launcher
// ---------------------------------------------------------------------------
extern "C" void kernel_launch(void* const* d_in, const int* in_sizes, int n_in,
                              void* d_out, int out_size, void* d_ws, size_t ws_size,
                              hipStream_t stream) {
    const float* img     = (const float*)d_in[0];   // [16,512,96,96]
    const float* gps     = (const float*)d_in[1];   // [16,2]
    const int*   mem_idx = (const int*)  d_in[2];   // [16]
    const float* membank = (const float*)d_in[3];   // [10000,20,512]
    const float* om0     = (const float*)d_in[4];
    const float* b0      = (const float*)d_in[5];
    const float* om1     = (const float*)d_in[6];
    const float* b1      = (const float*)d_in[7];
    const float* om2     = (const float*)d_in[8];
    const float* b2      = (const float*)d_in[9];
    const float* w1      = (const float*)d_in[10];  // [512,1024]
    const float* bb1     = (const float*)d_in[11];
    const float* w2      = (const float*)d_in[12];  // [1024,1024]
    const float* bb2     = (const float*)d_in[13];
    const float* w3      = (const float*)d_in[14];  // [1024,512]
    const float* bb3     = (const float*)d_in[15];
    const float* ln_g    = (const float*)d_in[16];
    const float* ln_b    = (const float*)d_in[17];
    // d_in[18..21] = wq,bq,wk,bk are mathematically dead (softmax over kv_len==1)
    const float* wv      = (const float*)d_in[22];  // [512,512]
    const float* bv      = (const float*)d_in[23];
    float* out = (float*)d_out;

    // workspace layout (floats)
    float* ws   = (float*)d_ws;
    float* r    = ws;                 // [16,512]
    float* h1   = r    + 16 * 512;    // [16,1024]
    float* h2   = h1   + 16 * 1024;   // [16,1024]
    float* loc  = h2   + 16 * 1024;   // [16,512]
    float* locf = loc  + 16 * 512;    // [16,512]
    float* V    = locf + 16 * 512;    // [16,512]

    rff_kernel<<<16, 512, 0, stream>>>(gps, om0, b0, om1, b1, om2, b2, r);

    gemm16_wmma<<<1024 / 16, 32, 0, stream>>>(r,  w1, bb1, h1,  512, 1024, 1);
    gemm16_wmma<<<1024 / 16, 32, 0, stream>>>(h1, w2, bb2, h2, 1024, 1024, 1);
    gemm16_wmma<<< 512 / 16, 32, 0, stream>>>(h2, w3, bb3, loc, 1024, 512, 0);

    ln_blend<<<16, 512, 0, stream>>>(loc, ln_g, ln_b, membank, mem_idx, locf);

    gemm16_wmma<<< 512 / 16, 32, 0, stream>>>(locf, wv, bv, V, 512, 512, 0);

    fuse_add<<<16 * 512, 256, 0, stream>>>(img, V, out);
}